// DistanceLoss_20564303414098
// MI455X (gfx1250) — compile-verified
//
#include <hip/hip_runtime.h>
#include <hip/hip_bf16.h>
#include <math.h>

typedef __attribute__((ext_vector_type(2))) float v2f;
typedef __attribute__((ext_vector_type(8))) float v8f;

#define B_  4
#define C_  8
#define H_  512
#define W_  512
#define HW  (H_ * W_)
#define L_  7
#define CAP 51   // min(distance, 51): boundary interior pixel = 1, 50 iters + mask

#if __has_builtin(__builtin_amdgcn_global_load_async_to_lds_b128)
typedef int v4i_t __attribute__((vector_size(16)));
typedef __attribute__((address_space(1))) v4i_t* async_gp_t;   // global int4*
typedef __attribute__((address_space(3))) v4i_t* async_lp_t;   // LDS int4*
#endif

// ---------------------------------------------------------------------------
// ws layout:
//   [  0, 32)  float    sums[8]   (per-class sum of norm_diff)
//   [ 32, 64)  unsigned pcnt[8]   (per-class count of pred pixels)
//   [ 64, 96)  unsigned tcnt[8]   (per-class count of target pixels)
//   [128, 128+1MB)         u8 pred[B,H,W]
//   [128+1MB, +7MB)        u8 g_in [L,B,H,W]  vertical dist-to-background
//   [.. , +7MB)            u8 g_out[L,B,H,W]  vertical dist-to-mask
// ---------------------------------------------------------------------------

__global__ void k_zero(unsigned* acc) {
    int i = threadIdx.x;
    if (i < 24) acc[i] = 0u;
}

// argmax over the 8 class logits, 4 pixels per thread via b128 loads.
__global__ void k_argmax(const float4* __restrict__ logits,
                         uchar4* __restrict__ pred) {
    int idx = blockIdx.x * blockDim.x + threadIdx.x;   // 0 .. B*HW/4-1
    int b  = idx >> 16;                                // HW/4 = 65536
    int p4 = idx & 65535;
    const float4* base = logits + (size_t)b * (C_ * HW / 4) + p4;
    float4 bv = base[0];
    int cx = 0, cy = 0, cz = 0, cw = 0;
#pragma unroll
    for (int c = 1; c < C_; ++c) {
        float4 v = base[(size_t)c * (HW / 4)];
        if (v.x > bv.x) { bv.x = v.x; cx = c; }   // first-max (jnp.argmax)
        if (v.y > bv.y) { bv.y = v.y; cy = c; }
        if (v.z > bv.z) { bv.z = v.z; cz = c; }
        if (v.w > bv.w) { bv.w = v.w; cw = c; }
    }
    pred[idx] = make_uchar4((unsigned char)cx, (unsigned char)cy,
                            (unsigned char)cz, (unsigned char)cw);
}

// Vertical 1D distance transform per (label, batch, column), clamped at CAP.
// Image border is neither background (erosion pads +inf) nor mask (dilation
// pads -inf), so both scans start from CAP at the edges.
__global__ void k_vdt(const int* __restrict__ target,
                      unsigned char* __restrict__ gin,
                      unsigned char* __restrict__ gout) {
    int idx = blockIdx.x * blockDim.x + threadIdx.x;   // L*B*W = 14336
    if (idx >= L_ * B_ * W_) return;
    int l   = idx / (B_ * W_);
    int rem = idx - l * (B_ * W_);
    int b   = rem / W_;
    int x   = rem - b * W_;
    int cls = l + 1;

    const int* tcol = target + (size_t)b * HW + x;
    unsigned char* gi = gin  + (size_t)(l * B_ + b) * HW + x;
    unsigned char* go = gout + (size_t)(l * B_ + b) * HW + x;

    int pin = CAP, pout = CAP;                 // down pass
    for (int y = 0; y < H_; ++y) {
        if ((y & 7) == 0 && y + 8 < H_)        // global_prefetch_b8 path
            __builtin_prefetch(tcol + (size_t)(y + 8) * W_, 0, 3);
        int t = (tcol[(size_t)y * W_] == cls);
        pin  = t ? min(pin + 1, CAP) : 0;      // dist to background (tgt != c)
        pout = t ? 0 : min(pout + 1, CAP);     // dist to mask       (tgt == c)
        gi[(size_t)y * W_] = (unsigned char)pin;
        go[(size_t)y * W_] = (unsigned char)pout;
    }
    int cin = gi[(size_t)(H_ - 1) * W_];       // up pass
    int cot = go[(size_t)(H_ - 1) * W_];
    for (int y = H_ - 2; y >= 0; --y) {
        int vi = min((int)gi[(size_t)y * W_], min(cin + 1, CAP));
        int vo = min((int)go[(size_t)y * W_], min(cot + 1, CAP));
        gi[(size_t)y * W_] = (unsigned char)vi;
        go[(size_t)y * W_] = (unsigned char)vo;
        cin = vi; cot = vo;
    }
}

// d(x) = min_dx max(|dx|, g[x+dx]) with early exit; exact for L-inf metric.
__device__ __forceinline__ int scan_row(const unsigned char* g, int x) {
    int best = g[x];
#pragma unroll 1
    for (int dx = 1; dx < CAP; ++dx) {
        if (dx >= best) break;                 // all further candidates >= dx
        int cand = 1000;
        int lo = x - dx, hi = x + dx;
        if (lo >= 0) cand = g[lo];
        if (hi < W_) cand = min(cand, (int)g[hi]);
        best = min(best, max(cand, dx));
    }
    return min(best, CAP);
}

// Horizontal DT pass + loss, one block per (label, batch, row).
__global__ void __launch_bounds__(256)
k_hdt_loss(const int* __restrict__ target,
           const unsigned char* __restrict__ pred,
           const unsigned char* __restrict__ gin,
           const unsigned char* __restrict__ gout,
           float* __restrict__ sums,
           unsigned* __restrict__ pcnt,
           unsigned* __restrict__ tcnt) {
    __shared__ __align__(16) unsigned char gin_s[W_];
    __shared__ __align__(16) unsigned char gout_s[W_];
    __shared__ float bsum;
    __shared__ unsigned bp, bt;

    int tid = threadIdx.x;
    int y = blockIdx.x, b = blockIdx.y, l = blockIdx.z;
    int cls = l + 1;
    size_t grow = ((size_t)(l * B_ + b) * H_ + y) * W_;
    size_t trow = (size_t)b * HW + (size_t)y * W_;

    if (tid == 0) { bsum = 0.0f; bp = 0u; bt = 0u; }

    // Stage the two 512B distance rows into LDS.
#if __has_builtin(__builtin_amdgcn_global_load_async_to_lds_b128)
    // CDNA5 async DMA: 32 lanes x 16B per row, no VGPR round-trip (ASYNCcnt).
    if (tid < 32) {
        __builtin_amdgcn_global_load_async_to_lds_b128(
            (async_gp_t)(gin + grow + (size_t)tid * 16),
            (async_lp_t)(gin_s + tid * 16), 0, 0);
        __builtin_amdgcn_global_load_async_to_lds_b128(
            (async_gp_t)(gout + grow + (size_t)tid * 16),
            (async_lp_t)(gout_s + tid * 16), 0, 0);
    }
#if __has_builtin(__builtin_amdgcn_s_wait_asynccnt)
    __builtin_amdgcn_s_wait_asynccnt(0);
#else
    asm volatile("s_wait_asynccnt 0" ::: "memory");
#endif
#else
    for (int i = tid; i < W_; i += 256) {
        gin_s[i]  = gin[grow + i];
        gout_s[i] = gout[grow + i];
    }
#endif
    __syncthreads();

    const float invL = 1.0f / log1pf(50.0f);   // 1 / log1p(MAX_ITER)

    v8f acc = {};                              // f32 16x16 WMMA accumulator
    v2f ones; ones.x = 1.0f; ones.y = 1.0f;    // all-ones B (layout-invariant)
    int ps = 0, ts = 0;

#pragma unroll
    for (int it = 0; it < 2; ++it) {
        int x = tid + it * 256;
        int t = (target[trow + x] == cls);
        int s = ((int)pred[trow + x] == cls);
        ps += s; ts += t;
        int diff;
        if (t) {                               // inside mask: d_pos = min(d,51)
            int d = scan_row(gin_s, x);
            diff = s ? 0 : d;                  // |s*d_pos - d_pos|
        } else {                               // outside: full_dist = -e (e<=50)
            int e = scan_row(gout_s, x);
            diff = (s && e <= 50) ? e : 0;     // |s*(-e) - 0|
        }
        float nrm = log1pf((float)diff + 1e-6f) * invL;

        // D[m][n] = sum_k A[m][k]*1 + C[m][n]; A holds this wave's 32 norms.
        v2f a; a.x = nrm; a.y = 0.0f;
        acc = __builtin_amdgcn_wmma_f32_16x16x4_f32(
                  /*neg_a=*/false, a, /*neg_b=*/false, ones,
                  /*c_mod=*/(short)0, acc, /*reuse_a=*/false, /*reuse_b=*/false);
    }

    // Column 0 of D: lane0 vgprs = M0..7, lane16 vgprs = M8..15 -> wave total.
    float colsum = acc[0] + acc[1] + acc[2] + acc[3] +
                   acc[4] + acc[5] + acc[6] + acc[7];
    float wtot = __shfl(colsum, 0, 32) + __shfl(colsum, 16, 32);
    if ((tid & 31) == 0) atomicAdd(&bsum, wtot);
    atomicAdd(&bp, (unsigned)ps);
    atomicAdd(&bt, (unsigned)ts);
    __syncthreads();

    if (tid == 0) {
        atomicAdd(&sums[l], bsum);
        atomicAdd(&pcnt[l], bp);
        atomicAdd(&tcnt[l], bt);
    }
}

__global__ void k_final(const float* __restrict__ sums,
                        const unsigned* __restrict__ pcnt,
                        const unsigned* __restrict__ tcnt,
                        float* __restrict__ out) {
    float total = 0.0f, count = 0.0f;
    const float denom = (float)(B_ * HW);
    for (int l = 0; l < L_; ++l) {
        float pcl   = sums[l] / denom;                       // per-class mean
        float valid = ((pcnt[l] > 0u) || (tcnt[l] > 0u)) ? 1.0f : 0.0f;
        total += pcl * valid;                                // weights all 1.0
        count += valid;
    }
    out[0] = (count > 0.0f) ? (total / fmaxf(count, 1.0f)) : 0.0f;
}

extern "C" void kernel_launch(void* const* d_in, const int* in_sizes, int n_in,
                              void* d_out, int out_size, void* d_ws, size_t ws_size,
                              hipStream_t stream) {
    (void)in_sizes; (void)n_in; (void)out_size; (void)ws_size;
    const float* logits = (const float*)d_in[0];
    const int*   target = (const int*)d_in[1];

    char* ws = (char*)d_ws;
    float*    sums = (float*)ws;
    unsigned* pcnt = (unsigned*)(ws + 32);
    unsigned* tcnt = (unsigned*)(ws + 64);
    unsigned char* pred = (unsigned char*)(ws + 128);
    unsigned char* gin  = pred + (size_t)B_ * HW;            // 1 MB
    unsigned char* gout = gin + (size_t)L_ * B_ * HW;        // 7 MB each

    k_zero<<<1, 32, 0, stream>>>((unsigned*)ws);
    k_argmax<<<(B_ * HW / 4) / 256, 256, 0, stream>>>((const float4*)logits,
                                                      (uchar4*)pred);
    k_vdt<<<(L_ * B_ * W_ + 255) / 256, 256, 0, stream>>>(target, gin, gout);
    dim3 g3(H_, B_, L_);
    k_hdt_loss<<<g3, 256, 0, stream>>>(target, pred, gin, gout, sums, pcnt, tcnt);
    k_final<<<1, 1, 0, stream>>>(sums, pcnt, tcnt, (float*)d_out);
}